// FeatureLoss_64037962383465
// MI455X (gfx1250) — compile-verified
//
#include <hip/hip_runtime.h>

// ---------------------------------------------------------------------------
// FeatureLoss: L1( gram(norm(out0)), gram(norm(tgt0)) ) + same for pair 1
//   x: [B=8, C=128, 48,48] -> fm [8,128,2304];  S = (fm/||fm||_col)^T (fm/..)
// Fused tiled Gram-difference, fp32 WMMA (V_WMMA_F32_16X16X4_F32), wave32.
// 128x128 macro tiles, triangular symmetry, channel-pair-interleaved LDS so
// every WMMA operand is one ds_load_b64 (no operand-assembly movs).
// ---------------------------------------------------------------------------

#define N_SP   2304     // 48*48
#define C_CH   128
#define BATCH  8
#define T128   18       // 2304 / 128
#define NPAIRS 171      // T128*(T128+1)/2  (ti <= tj)
#define SJ     288      // LDS panel pair-row stride (128*2 + 32 pad)
                        //  -> half-waves (pairRow+1) are 288%64=32 banks apart
#define PSZ    (8 * SJ) // one panel: 8 channel-pairs x 128 cols interleaved
#define EPS_N  1e-8f

typedef __attribute__((ext_vector_type(2))) float v2f;
typedef __attribute__((ext_vector_type(8))) float v8f;

#define WMMA_F32(A, B, Cacc) \
  __builtin_amdgcn_wmma_f32_16x16x4_f32(false, (A), false, (B), (short)0, (Cacc), false, false)

// --------------------------- kernel 1: inverse column norms -----------------
__global__ __launch_bounds__(256)
void colnorm_kernel(const float* __restrict__ x0, const float* __restrict__ x1,
                    const float* __restrict__ x2, const float* __restrict__ x3,
                    float* __restrict__ invn) {
  int gid = blockIdx.x * 256 + threadIdx.x;          // [0, 4*8*2304)
  if (gid >= 4 * BATCH * N_SP) return;
  int n = gid % N_SP;
  int b = (gid / N_SP) % BATCH;
  int t = gid / (N_SP * BATCH);
  const float* x = (t == 0) ? x0 : (t == 1) ? x1 : (t == 2) ? x2 : x3;
  const float* p = x + (size_t)b * C_CH * N_SP + n;
  float s = 0.f;
#pragma unroll 8
  for (int c = 0; c < C_CH; ++c) {
    float v = p[(size_t)c * N_SP];
    s += v * v;
  }
  invn[gid] = 1.0f / (sqrtf(s) + EPS_N);
}

// --------------------- kernel 2: fused Gram-diff L1 (WMMA) ------------------
// grid: (NPAIRS, 16)  blockIdx.y = p*8 + b ; block = 256 thr = 8 wave32
__global__ __launch_bounds__(256)
void gram_l1_kernel(const float* __restrict__ o0, const float* __restrict__ o1,
                    const float* __restrict__ g0, const float* __restrict__ g1,
                    const float* __restrict__ invn, float* __restrict__ partials) {
  __shared__ float smem[4 * PSZ];         // panels: [out-i][out-j][tgt-i][tgt-j]
  __shared__ float sInv[4][128];          // staged inverse norms
  __shared__ float sRed[256];

  const int tid  = threadIdx.x;
  const int lane = tid & 31;
  const int wave = tid >> 5;
  const int l16  = lane & 15;
  const int hi   = lane >> 4;             // half-wave -> +1 channel pair

  const int pb = blockIdx.y;
  const int p  = pb >> 3;                 // pair 0/1
  const int b  = pb & 7;                  // batch

  // decode triangular (ti <= tj) job id
  int pair = blockIdx.x;
  int ti = 0;
  while (pair >= (T128 - ti)) { pair -= (T128 - ti); ++ti; }
  const int tj = ti + pair;
  const int i0 = ti * 128, j0 = tj * 128;
  const float wsym = (ti == tj) ? 1.0f : 2.0f;

  const float* Xo = (p == 0) ? o0 : o1;
  const float* Xg = (p == 0) ? g0 : g1;
  const size_t boff = (size_t)b * C_CH * N_SP;
  const float* invO = invn + (size_t)(p * BATCH + b) * N_SP;
  const float* invG = invn + (size_t)((p + 2) * BATCH + b) * N_SP;

  // stage the 4 x 128 inverse-norm slices
  for (int e = tid; e < 512; e += 256) {
    int a = e >> 7, off = e & 127;
    const float* src = (a == 0) ? (invO + i0) : (a == 1) ? (invO + j0)
                     : (a == 2) ? (invG + i0) : (invG + j0);
    sInv[a][off] = src[off];
  }

  // wave -> 32x64 sub-block: 2 i-subtiles x 4 j-subtiles
  const int iw = wave & 3;                // rows [iw*32, +32)
  const int jw = wave >> 2;               // cols [jw*64, +64)
  const int rb0 = iw * 32, rb1 = iw * 32 + 16;
  const int cbase = jw * 64;

  const v8f vzero = {0.f, 0.f, 0.f, 0.f, 0.f, 0.f, 0.f, 0.f};
  v8f accO[2][4], accG[2][4];
#pragma unroll
  for (int ii = 0; ii < 2; ++ii)
#pragma unroll
    for (int jj = 0; jj < 4; ++jj) { accO[ii][jj] = vzero; accG[ii][jj] = vzero; }

  for (int c0 = 0; c0 < C_CH; c0 += 16) {
    __syncthreads();
    // fill 4 panels: channels [c0,c0+16) as 8 interleaved pairs x 128 cols
    for (int job = tid; job < 1024; job += 256) {
      const int q     = job & 31;         // quad of 4 cols (coalesced fastest)
      const int pr    = (job >> 5) & 7;   // channel pair
      const int panel = job >> 8;
      const float* X  = (panel < 2) ? Xo : Xg;
      const int colb  = (panel & 1) ? j0 : i0;
      const float* gs = X + boff + (size_t)(c0 + 2 * pr) * N_SP + colb + 4 * q;
      const float4 v0 = *(const float4*)gs;
      const float4 v1 = *(const float4*)(gs + N_SP);
      float2* dst = (float2*)(smem + panel * PSZ + pr * SJ + 8 * q);
      dst[0] = make_float2(v0.x, v1.x);
      dst[1] = make_float2(v0.y, v1.y);
      dst[2] = make_float2(v0.z, v1.z);
      dst[3] = make_float2(v0.w, v1.w);
    }
    __syncthreads();

#pragma unroll
    for (int cc = 0; cc < 16; cc += 4) {
      const int prB = ((cc >> 1) + hi) * SJ;     // pairRow = cc/2 (+1 hi half)
      const float* Oi = smem + 0 * PSZ + prB;
      const float* Oj = smem + 1 * PSZ + prB;
      const float* Gi = smem + 2 * PSZ + prB;
      const float* Gj = smem + 3 * PSZ + prB;
      const v2f aO0 = *(const v2f*)(Oi + (rb0 + l16) * 2);
      const v2f aO1 = *(const v2f*)(Oi + (rb1 + l16) * 2);
      const v2f aG0 = *(const v2f*)(Gi + (rb0 + l16) * 2);
      const v2f aG1 = *(const v2f*)(Gi + (rb1 + l16) * 2);
#pragma unroll
      for (int jj = 0; jj < 4; ++jj) {
        const v2f bO = *(const v2f*)(Oj + (cbase + 16 * jj + l16) * 2);
        const v2f bG = *(const v2f*)(Gj + (cbase + 16 * jj + l16) * 2);
        accO[0][jj] = WMMA_F32(aO0, bO, accO[0][jj]);
        accO[1][jj] = WMMA_F32(aO1, bO, accO[1][jj]);
        accG[0][jj] = WMMA_F32(aG0, bG, accG[0][jj]);
        accG[1][jj] = WMMA_F32(aG1, bG, accG[1][jj]);
      }
    }
  }

  // epilogue: S = G * inv[row] * inv[col]; L1 of (S_out - S_tgt)
  const int rowHalf = hi * 8;             // C/D layout: VGPR v -> row v (+8 hi)
  float s = 0.f;
#pragma unroll
  for (int jj = 0; jj < 4; ++jj) {
    const int col = cbase + 16 * jj + l16;
    const float cO = sInv[1][col], cG = sInv[3][col];
#pragma unroll
    for (int ii = 0; ii < 2; ++ii) {
#pragma unroll
      for (int v = 0; v < 8; ++v) {
        const int row = iw * 32 + ii * 16 + rowHalf + v;
        const float rO = sInv[0][row], rG = sInv[2][row];
        s += fabsf(accO[ii][jj][v] * rO * cO - accG[ii][jj][v] * rG * cG);
      }
    }
  }
  s *= wsym;

  sRed[tid] = s;
  __syncthreads();
  for (int st = 128; st > 0; st >>= 1) {
    if (tid < st) sRed[tid] += sRed[tid + st];
    __syncthreads();
  }
  if (tid == 0) partials[(size_t)pb * NPAIRS + blockIdx.x] = sRed[0];
}

// ------------------------ kernel 3: deterministic finalize ------------------
__global__ __launch_bounds__(256)
void finalize_kernel(const float* __restrict__ partials, float* __restrict__ out) {
  __shared__ float sRed[256];
  const int tid = threadIdx.x;
  float s = 0.f;
  for (int i = tid; i < 16 * NPAIRS; i += 256) s += partials[i];
  sRed[tid] = s;
  __syncthreads();
  for (int st = 128; st > 0; st >>= 1) {
    if (tid < st) sRed[tid] += sRed[tid + st];
    __syncthreads();
  }
  if (tid == 0) out[0] = sRed[0] * (1.0f / (8.0f * 2304.0f * 2304.0f));
}

// ---------------------------------------------------------------------------
extern "C" void kernel_launch(void* const* d_in, const int* in_sizes, int n_in,
                              void* d_out, int out_size, void* d_ws, size_t ws_size,
                              hipStream_t stream) {
  const float* o0 = (const float*)d_in[0];
  const float* o1 = (const float*)d_in[1];
  const float* g0 = (const float*)d_in[2];
  const float* g1 = (const float*)d_in[3];

  float* invn     = (float*)d_ws;                 // 4*8*2304 floats (~295 KB)
  float* partials = invn + 4 * BATCH * N_SP;      // 16*171 floats

  dim3 gridN((4 * BATCH * N_SP + 255) / 256);
  colnorm_kernel<<<gridN, 256, 0, stream>>>(o0, o1, g0, g1, invn);

  dim3 gridG(NPAIRS, 16);
  gram_l1_kernel<<<gridG, 256, 0, stream>>>(o0, o1, g0, g1, invn, partials);

  finalize_kernel<<<1, 256, 0, stream>>>(partials, (float*)d_out);
}